// ChemiNet_53283364274341
// MI455X (gfx1250) — compile-verified
//
#include <hip/hip_runtime.h>
#include <hip/hip_bf16.h>
#include <cstddef>

// ---------------- problem constants (match reference) ----------------
#define NN   20000   // nodes
#define EE   40000   // edges
#define FA   75      // atom features
#define FB   12      // bond features
#define CC   100     // output channels
#define GG   1000    // molecules
#define NT_O 7       // 7 column tiles of 16 cover C=100 (padded to 112)
#define EPS  1e-5f

typedef float v2f __attribute__((ext_vector_type(2)));
typedef float v8f __attribute__((ext_vector_type(8)));

// D = A(16x4 f32) * B(4x16 f32) + C(16x16 f32)
static __device__ inline v8f wmma_f32_16x16x4(v2f a, v2f b, v8f c) {
  return __builtin_amdgcn_wmma_f32_16x16x4_f32(
      /*neg_a=*/false, a, /*neg_b=*/false, b,
      /*c_mod=*/(short)0, c, /*reuse_a=*/false, /*reuse_b=*/false);
}

static __device__ inline float relu1(float v) {
  // single v_max_num_f32 (avoids the canonicalize+max pair from ?:)
  return __builtin_fmaxf(v, 0.f);
}

// ---------------------------------------------------------------------
// K0: zero scratch
// ---------------------------------------------------------------------
__global__ void zero_kernel(float* __restrict__ p, int n) {
  int i = blockIdx.x * 256 + threadIdx.x;
  if (i < n) p[i] = 0.f;
}

// ---------------------------------------------------------------------
// K1: fused NNConv edge messages.
// Per wave: 16 edges. ew tile [16 x 16] lives only in registers (v8f acc),
// computed as 3x WMMA f32 16x16x4 over K=12, + b1, relu, scaled by
// x[src,i] (LDS broadcast), accumulated into msg frags over i=0..74,
// then scattered into agg with global f32 atomics.
// Grid: 625 blocks x 128 threads (4 waves) -> 40000 edges exactly.
// ---------------------------------------------------------------------
__global__ __launch_bounds__(128)
void edge_msg_kernel(const float* __restrict__ x,
                     const float* __restrict__ edge_attr,
                     const float* __restrict__ W1,
                     const float* __restrict__ b1,
                     const int*   __restrict__ edge_index,
                     float*       __restrict__ agg)
{
  __shared__ float lds_x[64 * 76];   // x[src] rows for this block's 64 edges
  const int tid  = threadIdx.x;
  const int wave = tid >> 5;
  const int lane = tid & 31;
  const int ebase_blk = blockIdx.x * 64;
  const int* __restrict__ srcs = edge_index;        // edge_index[0]
  const int* __restrict__ dsts = edge_index + EE;   // edge_index[1]

  // cooperatively gather x[src[e], :] into LDS (64 x 75, stride 76)
  for (int idx = tid; idx < 64 * FA; idx += 128) {
    int e = idx / FA;
    int f = idx - e * FA;
    int s = srcs[ebase_blk + e];
    lds_x[e * 76 + f] = x[(size_t)s * FA + f];
  }
  __syncthreads();

  const int ebase = ebase_blk + wave * 16;
  const int lrow  = lane & 15;     // A: row m, B/C/D: column n
  const int khalf = lane >> 4;     // 0: K={0,1} half, 1: K={2,3} half
  const int mrow  = khalf ? 8 : 0; // C/D rows covered by this lane half

  // A fragments: edge_attr tile [16 x 12]; frag ks covers K = ks*4 .. ks*4+3
  v2f afrag[3];
  {
    const float* ea = edge_attr + (size_t)(ebase + lrow) * FB;
#pragma unroll
    for (int ks = 0; ks < 3; ++ks) {
      int k0 = ks * 4 + khalf * 2;
      afrag[ks].x = ea[k0];
      afrag[ks].y = ea[k0 + 1];
    }
  }

  // message accumulators: 7 column tiles (= 56 VGPRs)
  v8f msg[NT_O];
#pragma unroll
  for (int t = 0; t < NT_O; ++t)
#pragma unroll
    for (int r = 0; r < 8; ++r) msg[t][r] = 0.f;

  const int xrow_base = (wave * 16 + mrow) * 76;

  for (int i = 0; i < FA; ++i) {
    // per-row scales x[src[edge m], i]  (uniform per lane-half -> LDS bcast)
    float xs[8];
#pragma unroll
    for (int r = 0; r < 8; ++r) xs[r] = lds_x[xrow_base + r * 76 + i];

#pragma unroll
    for (int t = 0; t < NT_O; ++t) {
      int  col = t * 16 + lrow;            // 0..111; valid if < 100
      bool okc = col < CC;
      float bval = okc ? b1[i * CC + col] : 0.f;
      v8f acc;
#pragma unroll
      for (int r = 0; r < 8; ++r) acc[r] = bval;

#pragma unroll
      for (int ks = 0; ks < 3; ++ks) {
        int    k0   = ks * 4 + khalf * 2;
        size_t base = (size_t)k0 * (FA * CC) + (size_t)i * CC + col;
        v2f bfrag;
        bfrag.x = okc ? W1[base] : 0.f;
        bfrag.y = okc ? W1[base + (size_t)FA * CC] : 0.f;
        acc = wmma_f32_16x16x4(afrag[ks], bfrag, acc);
      }
      // relu then scale-accumulate into message
#pragma unroll
      for (int r = 0; r < 8; ++r)
        msg[t][r] = __builtin_fmaf(relu1(acc[r]), xs[r], msg[t][r]);
    }
  }

  // scatter-add messages into agg[dst, col]
#pragma unroll
  for (int t = 0; t < NT_O; ++t) {
    int col = t * 16 + lrow;
    if (col < CC) {
#pragma unroll
      for (int r = 0; r < 8; ++r) {
        int e = ebase + mrow + r;
        int d = dsts[e];
        unsafeAtomicAdd(&agg[(size_t)d * CC + col], msg[t][r]);
      }
    }
  }
}

// ---------------------------------------------------------------------
// K2: h = relu(x @ W_root + agg + bias), overwrite agg in place,
//     accumulate per-channel sum / sumsq for BatchNorm.
// Per wave: one 16-node tile, all 7 column tiles; K = 75 padded to 76
// (19 WMMA K-steps). 1250 tiles = 313 blocks x 4 waves (last 2 idle).
// ---------------------------------------------------------------------
__global__ __launch_bounds__(128)
void node_kernel(const float* __restrict__ x,
                 const float* __restrict__ W_root,
                 const float* __restrict__ bias,
                 float*       __restrict__ aggh,
                 float*       __restrict__ sums,
                 float*       __restrict__ sumsq)
{
  const int tid  = threadIdx.x;
  const int wave = tid >> 5;
  const int lane = tid & 31;
  const int tile = blockIdx.x * 4 + wave;
  if (tile >= NN / 16) return;            // wave-uniform exit

  const int nb    = tile * 16;
  const int lrow  = lane & 15;
  const int khalf = lane >> 4;
  const int mrow  = khalf ? 8 : 0;

  // A frags: x[nb+lrow, k], K padded 75 -> 76
  v2f a[19];
  const float* xr = x + (size_t)(nb + lrow) * FA;
#pragma unroll
  for (int ks = 0; ks < 19; ++ks) {
    int k0 = ks * 4 + khalf * 2;
    a[ks].x = (k0     < FA) ? xr[k0]     : 0.f;
    a[ks].y = (k0 + 1 < FA) ? xr[k0 + 1] : 0.f;
  }

#pragma unroll
  for (int t = 0; t < NT_O; ++t) {
    int  col = t * 16 + lrow;
    bool okc = col < CC;

    v8f acc;
#pragma unroll
    for (int r = 0; r < 8; ++r)
      acc[r] = okc ? (aggh[(size_t)(nb + mrow + r) * CC + col] + bias[col]) : 0.f;

#pragma unroll
    for (int ks = 0; ks < 19; ++ks) {
      int k0 = ks * 4 + khalf * 2;
      v2f b;
      b.x = (okc && k0     < FA) ? W_root[(size_t)k0       * CC + col] : 0.f;
      b.y = (okc && k0 + 1 < FA) ? W_root[(size_t)(k0 + 1) * CC + col] : 0.f;
      acc = wmma_f32_16x16x4(a[ks], b, acc);
    }

    float p = 0.f, psq = 0.f;
#pragma unroll
    for (int r = 0; r < 8; ++r) {
      float h = relu1(acc[r]);
      if (okc) aggh[(size_t)(nb + mrow + r) * CC + col] = h;
      p += h;
      psq = __builtin_fmaf(h, h, psq);
    }
    // fold lanes 16-31 (rows 8-15) into lanes 0-15 (same column)
    p   += __shfl_down(p,   16, 32);
    psq += __shfl_down(psq, 16, 32);
    if (okc && lane < 16) {
      unsafeAtomicAdd(&sums[col],  p);
      unsafeAtomicAdd(&sumsq[col], psq);
    }
  }
}

// ---------------------------------------------------------------------
// K3: per-channel BatchNorm scale/shift from batch statistics
// ---------------------------------------------------------------------
__global__ void stats_kernel(const float* __restrict__ sums,
                             const float* __restrict__ sumsq,
                             const float* __restrict__ gamma,
                             const float* __restrict__ beta,
                             float* __restrict__ scale,
                             float* __restrict__ shift)
{
  int c = threadIdx.x;
  if (c < CC) {
    float inv_n = 1.0f / (float)NN;
    float mean  = sums[c] * inv_n;
    float var   = sumsq[c] * inv_n - mean * mean;
    float sc    = gamma[c] * rsqrtf(var + EPS);
    scale[c] = sc;
    shift[c] = beta[c] - mean * sc;
  }
}

// ---------------------------------------------------------------------
// K4: normalize h and segment-sum (global_add_pool) into pooled[G, C]
// ---------------------------------------------------------------------
__global__ void pool_kernel(const float* __restrict__ h,
                            const float* __restrict__ scale,
                            const float* __restrict__ shift,
                            const int*   __restrict__ batch,
                            float*       __restrict__ pooled)
{
  int idx = blockIdx.x * 256 + threadIdx.x;
  if (idx < NN * CC) {
    int n = idx / CC;
    int c = idx - n * CC;
    float v = __builtin_fmaf(h[idx], scale[c], shift[c]);
    unsafeAtomicAdd(&pooled[(size_t)batch[n] * CC + c], v);
  }
}

// ---------------------------------------------------------------------
// K5: out[g] = relu(pooled[g,:]) @ W_out + b_out
// ---------------------------------------------------------------------
__global__ void out_kernel(const float* __restrict__ pooled,
                           const float* __restrict__ W_out,
                           const float* __restrict__ b_out,
                           float*       __restrict__ out)
{
  int g = blockIdx.x * 128 + threadIdx.x;
  if (g < GG) {
    float acc = b_out[0];
    const float* row = pooled + (size_t)g * CC;
#pragma unroll 4
    for (int c = 0; c < CC; ++c)
      acc = __builtin_fmaf(relu1(row[c]), W_out[c], acc);
    out[g] = acc;
  }
}

// ---------------------------------------------------------------------
// launch: workspace layout (floats):
//   [0, 2'000'000)              agg -> h (in place)
//   [2'000'000, +128)           sums
//   [2'000'128, +128)           sumsq
//   [2'000'256, +128)           scale
//   [2'000'384, +128)           shift
//   [2'000'512, +100'000)       pooled
// total 2'100'512 floats = 8.4 MB
// ---------------------------------------------------------------------
extern "C" void kernel_launch(void* const* d_in, const int* in_sizes, int n_in,
                              void* d_out, int out_size, void* d_ws, size_t ws_size,
                              hipStream_t stream) {
  const float* x          = (const float*)d_in[0];
  const float* edge_attr  = (const float*)d_in[1];
  const float* W1         = (const float*)d_in[2];
  const float* b1         = (const float*)d_in[3];
  const float* W_root     = (const float*)d_in[4];
  const float* bias       = (const float*)d_in[5];
  const float* gamma      = (const float*)d_in[6];
  const float* beta       = (const float*)d_in[7];
  const float* W_out      = (const float*)d_in[8];
  const float* b_out      = (const float*)d_in[9];
  const int*   edge_index = (const int*)  d_in[10];
  const int*   batch      = (const int*)  d_in[11];
  float* out = (float*)d_out;

  float* W      = (float*)d_ws;
  float* agg    = W;                       // also h after node_kernel
  float* sums   = W + 2000000;
  float* sumsq  = sums  + 128;
  float* scale  = sumsq + 128;
  float* shift  = scale + 128;
  float* pooled = shift + 128;

  const int ztotal = 2100512;
  zero_kernel<<<(ztotal + 255) / 256, 256, 0, stream>>>(W, ztotal);

  edge_msg_kernel<<<EE / 64, 128, 0, stream>>>(x, edge_attr, W1, b1,
                                               edge_index, agg);

  node_kernel<<<(NN / 16 + 3) / 4, 128, 0, stream>>>(x, W_root, bias,
                                                     agg, sums, sumsq);

  stats_kernel<<<1, 128, 0, stream>>>(sums, sumsq, gamma, beta, scale, shift);

  pool_kernel<<<(NN * CC + 255) / 256, 256, 0, stream>>>(agg, scale, shift,
                                                         batch, pooled);

  out_kernel<<<(GG + 127) / 128, 128, 0, stream>>>(pooled, W_out, b_out, out);
}